// GNNEncoder_80530636800004
// MI455X (gfx1250) — compile-verified
//
#include <hip/hip_runtime.h>
#include <hip/hip_bf16.h>
#include <math.h>

typedef __attribute__((ext_vector_type(2))) float v2f;
typedef __attribute__((ext_vector_type(8))) float v8f;
typedef int v4i_vs __attribute__((vector_size(16)));   // matches async-LDS builtin param type

#define HID 128
#define NGRAPH 64
#define LDSW_STRIDE 136   // 128 + 8 pad -> the two 16-lane halves hit disjoint LDS bank windows

#define AS1 __attribute__((address_space(1)))
#define AS3 __attribute__((address_space(3)))

// ---------------------------------------------------------------------------
// Degree / normalization kernels
// ---------------------------------------------------------------------------
__global__ void deg_init_kernel(float* __restrict__ deg, int n) {
    int i = blockIdx.x * blockDim.x + threadIdx.x;
    if (i < n) deg[i] = 1.0f;  // self-loop contributes 1
}

__global__ void deg_count_kernel(const int* __restrict__ col, float* __restrict__ deg, int E) {
    int e = blockIdx.x * blockDim.x + threadIdx.x;
    if (e < E) unsafeAtomicAdd(&deg[col[e]], 1.0f);
}

__global__ void dis_kernel(const float* __restrict__ deg, float* __restrict__ dis, int n) {
    int i = blockIdx.x * blockDim.x + threadIdx.x;
    if (i < n) dis[i] = rsqrtf(deg[i]);  // deg >= 1 always (self-loops)
}

// ---------------------------------------------------------------------------
// WMMA fp32 GEMM: C[M x 128] = A[M x 128] * W[128 x 128]
// One wave per 16-row strip; strip covers all 8 column tiles (reuse A frag).
// W is staged into LDS in two 64-row chunks (34.8 KB each) so the 8 waves of
// the block share one copy instead of each re-reading 64 KB from global.
// Staging uses GLOBAL_LOAD_ASYNC_TO_LDS_B128 (ASYNCcnt) when available.
// V_WMMA_F32_16X16X4_F32 layouts (ISA 7.12.2):
//   A 16x4:  lanes 0-15 -> {K=0,K=1}, lanes 16-31 -> {K=2,K=3}, M = lane&15
//   B 4x16:  v0 = row K (lanes0-15: K, lanes16-31: K+2), v1 = next row, N = lane&15
//   C/D:     VGPR j: lanes0-15 -> M=j, lanes16-31 -> M=j+8, N = lane&15
// ---------------------------------------------------------------------------
__global__ __launch_bounds__(256) void gemm_n128_kernel(const float* __restrict__ A,
                                                        const float* __restrict__ W,
                                                        float* __restrict__ C, int M) {
    __shared__ float lw[64 * LDSW_STRIDE];

    const int gwave  = (blockIdx.x * blockDim.x + threadIdx.x) >> 5;
    const int lane   = threadIdx.x & 31;
    const int row0   = gwave * 16;
    const bool active = (row0 < M);        // wave-uniform; EXEC all-1s inside
    const int half = lane >> 4;            // 0: lanes 0-15, 1: lanes 16-31
    const int l    = lane & 15;

    v8f acc[8];
    v8f zero = {};
    #pragma unroll
    for (int nt = 0; nt < 8; ++nt) acc[nt] = zero;

    const float* Arow = A + (size_t)(row0 + l) * HID;   // address math only if inactive

    for (int kc = 0; kc < 2; ++kc) {
        // ---- stage 64 rows of W into LDS (all waves participate) ----
        const float* Wc = W + (size_t)kc * 64 * HID;
#if __has_builtin(__builtin_amdgcn_global_load_async_to_lds_b128)
        #pragma unroll
        for (int i = 0; i < 8; ++i) {
            int cidx = threadIdx.x + i * 256;        // 64 rows * 32 16B-chunks
            int r = cidx >> 5, q = cidx & 31;
            __builtin_amdgcn_global_load_async_to_lds_b128(
                (AS1 v4i_vs*)(Wc + (size_t)r * HID + q * 4),
                (AS3 v4i_vs*)(lw + r * LDSW_STRIDE + q * 4),
                0, 0);
        }
        asm volatile("s_wait_asynccnt 0x0" ::: "memory");
#else
        #pragma unroll
        for (int i = 0; i < 8; ++i) {
            int cidx = threadIdx.x + i * 256;
            int r = cidx >> 5, q = cidx & 31;
            ((float4*)(lw + r * LDSW_STRIDE))[q] = ((const float4*)(Wc + (size_t)r * HID))[q];
        }
#endif
        __syncthreads();

        // ---- 64-deep K chunk of WMMAs, B fragments from LDS ----
        if (active) {
            const float* Ak = Arow + kc * 64;
            for (int k = 0; k < 64; k += 4) {
                v2f a;
                const float* ap = Ak + k + half * 2;
                a.x = ap[0];
                a.y = ap[1];
                #pragma unroll
                for (int nt = 0; nt < 8; ++nt) {
                    const int n = nt * 16 + l;
                    const float* bp = lw + (k + half * 2) * LDSW_STRIDE + n;
                    v2f b;
                    b.x = bp[0];
                    b.y = bp[LDSW_STRIDE];
                    acc[nt] = __builtin_amdgcn_wmma_f32_16x16x4_f32(
                        false, a, false, b, (short)0, acc[nt], false, false);
                }
            }
        }
        __syncthreads();   // before next chunk overwrites lw
    }

    if (active) {
        #pragma unroll
        for (int j = 0; j < 8; ++j) {
            const int r = row0 + j + half * 8;
            float* cp = C + (size_t)r * HID + l;
            #pragma unroll
            for (int nt = 0; nt < 8; ++nt) cp[nt * 16] = acc[nt][j];
        }
    }
}

// ---------------------------------------------------------------------------
// acc[i,:] = t[i,:] * dis[i]^2   (self-loop message)
// ---------------------------------------------------------------------------
__global__ void selfloop_init_kernel(const float* __restrict__ t, const float* __restrict__ dis,
                                     float* __restrict__ acc, int n) {
    size_t idx = (size_t)blockIdx.x * blockDim.x + threadIdx.x;   // over n*32 float4s
    if (idx >= (size_t)n * 32) return;
    int i = (int)(idx >> 5), q = (int)(idx & 31);
    float d = dis[i];
    float w = d * d;
    float4 v = ((const float4*)(t + (size_t)i * HID))[q];
    float4 o = make_float4(v.x * w, v.y * w, v.z * w, v.w * w);
    ((float4*)(acc + (size_t)i * HID))[q] = o;
}

// ---------------------------------------------------------------------------
// Edge scatter: acc[col[e],:] += t[row[e],:] * dis[row]*dis[col]
// One wave per edge; lane owns a float4 of the 128 features.
// t / acc are 51 MB each -> L2-resident on MI455X (192 MB L2), so the random
// gather + global_atomic_add_f32 scatter run at L2 rates, not HBM.
// ---------------------------------------------------------------------------
__global__ void edge_agg_kernel(const int* __restrict__ row, const int* __restrict__ col,
                                const float* __restrict__ dis, const float* __restrict__ t,
                                float* __restrict__ acc, int E) {
    int e = blockIdx.x * (blockDim.x >> 5) + (threadIdx.x >> 5);
    if (e >= E) return;
    int lane = threadIdx.x & 31;
    int r = row[e], c = col[e];
    float w = dis[r] * dis[c];
    float4 v = ((const float4*)(t + (size_t)r * HID))[lane];
    float* ap = acc + (size_t)c * HID + lane * 4;
    unsafeAtomicAdd(ap + 0, v.x * w);
    unsafeAtomicAdd(ap + 1, v.y * w);
    unsafeAtomicAdd(ap + 2, v.z * w);
    unsafeAtomicAdd(ap + 3, v.w * w);
}

// ---------------------------------------------------------------------------
// acc[i,f] = relu(acc[i,f] + b[f])
// ---------------------------------------------------------------------------
__global__ void bias_relu_kernel(float* __restrict__ acc, const float* __restrict__ b, int n) {
    size_t idx = (size_t)blockIdx.x * blockDim.x + threadIdx.x;
    if (idx >= (size_t)n * 32) return;
    int i = (int)(idx >> 5), q = (int)(idx & 31);
    float4 v  = ((float4*)(acc + (size_t)i * HID))[q];
    float4 bb = ((const float4*)b)[q];
    v.x = fmaxf(v.x + bb.x, 0.0f);
    v.y = fmaxf(v.y + bb.y, 0.0f);
    v.z = fmaxf(v.z + bb.z, 0.0f);
    v.w = fmaxf(v.w + bb.w, 0.0f);
    ((float4*)(acc + (size_t)i * HID))[q] = v;
}

// ---------------------------------------------------------------------------
// Zero the pooled-sum / count buffers
// ---------------------------------------------------------------------------
__global__ void zero_kernel(float* __restrict__ p, int n) {
    int i = blockIdx.x * blockDim.x + threadIdx.x;
    if (i < n) p[i] = 0.0f;
}

// ---------------------------------------------------------------------------
// Mean-pool stage 1: per-block accumulation into a full 64x128 LDS tile
// (32 KB of the 320 KB WGP LDS) using ds_add_f32 atomics, then one flush of
// global f32 atomics per block. batch is sorted so LDS contention is low.
// ---------------------------------------------------------------------------
#define POOL_CHUNK 512
__global__ void pool_accum_kernel(const float* __restrict__ h, const int* __restrict__ batch,
                                  float* __restrict__ psum, float* __restrict__ pcnt, int n) {
    __shared__ float ls[NGRAPH * HID];
    __shared__ float lc[NGRAPH];
    for (int i = threadIdx.x; i < NGRAPH * HID; i += blockDim.x) ls[i] = 0.0f;
    if (threadIdx.x < NGRAPH) lc[threadIdx.x] = 0.0f;
    __syncthreads();

    const int wave = threadIdx.x >> 5;   // 8 waves per 256-thread block
    const int lane = threadIdx.x & 31;
    const int base = blockIdx.x * POOL_CHUNK;
    const int end  = min(base + POOL_CHUNK, n);
    for (int i = base + wave; i < end; i += 8) {
        int g = batch[i];
        float4 v = ((const float4*)(h + (size_t)i * HID))[lane];
        float* lp = &ls[g * HID + lane * 4];
        atomicAdd(lp + 0, v.x);
        atomicAdd(lp + 1, v.y);
        atomicAdd(lp + 2, v.z);
        atomicAdd(lp + 3, v.w);
        if (lane == 0) atomicAdd(&lc[g], 1.0f);
    }
    __syncthreads();

    for (int i = threadIdx.x; i < NGRAPH * HID; i += blockDim.x)
        if (ls[i] != 0.0f) unsafeAtomicAdd(&psum[i], ls[i]);
    if (threadIdx.x < NGRAPH && lc[threadIdx.x] != 0.0f)
        unsafeAtomicAdd(&pcnt[threadIdx.x], lc[threadIdx.x]);
}

// ---------------------------------------------------------------------------
// out[64,128] = (psum / max(cnt,1)) @ Wp + bp, via fp32 WMMA.
// 4 waves, one 16-row strip each. Single block.
// ---------------------------------------------------------------------------
__global__ void pool_gemm_kernel(const float* __restrict__ psum, const float* __restrict__ pcnt,
                                 const float* __restrict__ Wp, const float* __restrict__ bp,
                                 float* __restrict__ out) {
    const int wave = threadIdx.x >> 5;   // 0..3
    const int lane = threadIdx.x & 31;
    const int half = lane >> 4;
    const int l    = lane & 15;
    const int row0 = wave * 16;
    const int arow = row0 + l;
    const float inv = 1.0f / fmaxf(pcnt[arow], 1.0f);

    v8f acc[8];
    v8f zero = {};
    #pragma unroll
    for (int nt = 0; nt < 8; ++nt) acc[nt] = zero;

    const float* Arow = psum + (size_t)arow * HID;
    for (int k = 0; k < HID; k += 4) {
        v2f a;
        const float* ap = Arow + k + half * 2;
        a.x = ap[0] * inv;
        a.y = ap[1] * inv;
        #pragma unroll
        for (int nt = 0; nt < 8; ++nt) {
            const int n = nt * 16 + l;
            const float* wp = Wp + (size_t)(k + half * 2) * HID + n;
            v2f b;
            b.x = wp[0];
            b.y = wp[HID];
            acc[nt] = __builtin_amdgcn_wmma_f32_16x16x4_f32(
                false, a, false, b, (short)0, acc[nt], false, false);
        }
    }
    #pragma unroll
    for (int j = 0; j < 8; ++j) {
        const int r = row0 + j + half * 8;
        #pragma unroll
        for (int nt = 0; nt < 8; ++nt) {
            const int n = nt * 16 + l;
            out[(size_t)r * HID + n] = acc[nt][j] + bp[n];
        }
    }
}

// ---------------------------------------------------------------------------
// Launcher
// ---------------------------------------------------------------------------
extern "C" void kernel_launch(void* const* d_in, const int* in_sizes, int n_in,
                              void* d_out, int out_size, void* d_ws, size_t ws_size,
                              hipStream_t stream) {
    const float* x     = (const float*)d_in[0];
    const int*   ei    = (const int*)d_in[1];   // [2, E] flat
    const int*   batch = (const int*)d_in[2];
    const float* W1 = (const float*)d_in[3];
    const float* b1 = (const float*)d_in[4];
    const float* W2 = (const float*)d_in[5];
    const float* b2 = (const float*)d_in[6];
    const float* Wp = (const float*)d_in[7];
    const float* bp = (const float*)d_in[8];
    float* out = (float*)d_out;

    const int N = in_sizes[0] / HID;   // 100000
    const int E = in_sizes[1] / 2;     // 1600000
    const int* rowp = ei;              // sources
    const int* colp = ei + E;          // targets

    // Workspace layout (floats): t0[N*128] | acc[N*128] | deg[N] | dis[N] | psum[64*128] | pcnt[64]
    float* ws   = (float*)d_ws;
    float* t0   = ws;
    float* acc  = t0 + (size_t)N * HID;
    float* deg  = acc + (size_t)N * HID;
    float* dis  = deg + N;
    float* psum = dis + N;
    float* pcnt = psum + NGRAPH * HID;

    const int T = 256;
    const int blkN    = (N + T - 1) / T;
    const int blkE    = (E + T - 1) / T;
    const int blkNF   = (int)(((size_t)N * 32 + T - 1) / T);          // node*float4 elementwise
    const int blkEdge = (E + (T / 32) - 1) / (T / 32);                // wave per edge
    const int strips  = (N + 15) / 16;
    const int blkGemm = (strips + (T / 32) - 1) / (T / 32);           // wave per 16-row strip
    const int blkPool = (N + POOL_CHUNK - 1) / POOL_CHUNK;

    // --- normalization: deg = 1 + in-degree; dis = rsqrt(deg) ---
    deg_init_kernel<<<blkN, T, 0, stream>>>(deg, N);
    deg_count_kernel<<<blkE, T, 0, stream>>>(colp, deg, E);
    dis_kernel<<<blkN, T, 0, stream>>>(deg, dis, N);

    // --- layer 1 ---
    gemm_n128_kernel<<<blkGemm, T, 0, stream>>>(x, W1, t0, N);
    selfloop_init_kernel<<<blkNF, T, 0, stream>>>(t0, dis, acc, N);
    edge_agg_kernel<<<blkEdge, T, 0, stream>>>(rowp, colp, dis, t0, acc, E);
    bias_relu_kernel<<<blkNF, T, 0, stream>>>(acc, b1, N);

    // --- layer 2 ---
    gemm_n128_kernel<<<blkGemm, T, 0, stream>>>(acc, W2, t0, N);
    selfloop_init_kernel<<<blkNF, T, 0, stream>>>(t0, dis, acc, N);
    edge_agg_kernel<<<blkEdge, T, 0, stream>>>(rowp, colp, dis, t0, acc, E);
    bias_relu_kernel<<<blkNF, T, 0, stream>>>(acc, b2, N);

    // --- mean pool + projection ---
    zero_kernel<<<(NGRAPH * HID + NGRAPH + T - 1) / T, T, 0, stream>>>(psum, NGRAPH * HID + NGRAPH);
    pool_accum_kernel<<<blkPool, T, 0, stream>>>(acc, batch, psum, pcnt, N);
    pool_gemm_kernel<<<1, 128, 0, stream>>>(psum, pcnt, Wp, bp, out);
}